// MultiHeadAttentionConv_39127152066598
// MI455X (gfx1250) — compile-verified
//
#include <hip/hip_runtime.h>

#define NN   100000
#define EE   1600000
#define HH   8
#define DIMM 128   // = H*C
#define EDIM 32

typedef float v2f __attribute__((ext_vector_type(2)));
typedef float v8f __attribute__((ext_vector_type(8)));

// ---------------------------------------------------------------------------
// Float atomic max via ordered-int trick (segmax init'd to -inf).
// ---------------------------------------------------------------------------
__device__ __forceinline__ void atomicMaxFloat(float* addr, float val) {
    if (val >= 0.0f)
        atomicMax((int*)addr, __float_as_int(val));
    else
        atomicMin((unsigned int*)addr, __float_as_uint(val));
}

// ---------------------------------------------------------------------------
// Init: out = 0, denom = 0, segmax = -inf
// ---------------------------------------------------------------------------
__global__ void init_kernel(float* __restrict__ out, float* __restrict__ segmax,
                            float* __restrict__ denom, int nOut, int nSeg) {
    int i = blockIdx.x * blockDim.x + threadIdx.x;
    if (i < nOut) out[i] = 0.0f;
    if (i < nSeg) { segmax[i] = -__builtin_inff(); denom[i] = 0.0f; }
}

// ---------------------------------------------------------------------------
// WMMA GEMM: Out[M,NOUT] = A[M,K] * B[NB,K]^T   (row-major everywhere)
// One wave32 per 16x16 output tile; K/NB/NOUT compile-time => full unroll,
// immediate load offsets, no EXEC predication in the hot loop (row pointers
// are clamped; out-of-range lanes only pollute rows/cols that are never
// stored, since D[i,j] depends only on A row i and B row j).
// ---------------------------------------------------------------------------
template <int K, int NB, int NOUT>
__global__ void gemm_xwT_wmma(const float* __restrict__ A,
                              const float* __restrict__ B,
                              float* __restrict__ Out,
                              int M) {
    const int lane = threadIdx.x & 31;
    const int wid  = blockIdx.x * (blockDim.x >> 5) + (threadIdx.x >> 5);
    const int tilesM = (M + 15) >> 4;
    constexpr int tilesN = (NOUT + 15) >> 4;
    if (wid >= tilesM * tilesN) return;
    const int tm = wid / tilesN;
    const int tn = wid % tilesN;

    const int half = lane >> 4;   // selects K-pair group (see 16x4 f32 layout)
    const int l15  = lane & 15;

    int arow = tm * 16 + l15;                 // A row handled by this lane
    int bcol = tn * 16 + l15;                 // B row (= output column)
    const int arow_c = arow < M ? arow : M - 1;      // clamp: keeps loads in-bounds
    const int bcol_c = bcol < NB ? bcol : NB - 1;
    const float* Arow = A + (size_t)arow_c * K + half * 2;
    const float* Brow = B + (size_t)bcol_c * K + half * 2;

    v8f c = {0.f, 0.f, 0.f, 0.f, 0.f, 0.f, 0.f, 0.f};

    #pragma unroll
    for (int k0 = 0; k0 < K; k0 += 4) {
        v2f a = *(const v2f*)(Arow + k0);     // A[row, k0+2h .. +1]
        v2f b = *(const v2f*)(Brow + k0);     // B^T[k0+2h .., col]
        c = __builtin_amdgcn_wmma_f32_16x16x4_f32(
                false, a, false, b, (short)0, c, false, false);
    }

    // C/D layout: VGPR r holds row (r + 8*half), column = l15.
    const int ocol = tn * 16 + l15;
    if (ocol < NOUT) {
        #pragma unroll
        for (int r = 0; r < 8; ++r) {
            const int orow = tm * 16 + r + 8 * half;
            if (orow < M) Out[(size_t)orow * NOUT + ocol] = c[r];
        }
    }
}

// ---------------------------------------------------------------------------
// Per-edge attention logits. One wave32 per edge; lane covers 4 channels of
// head (lane>>2). score[e,h] = dot(q[dst,h,:], k[src,h,:]) / 4 + bias[e,h].
// Also atomic-max into segmax[dst,h].
// ---------------------------------------------------------------------------
__global__ void edge_scores_kernel(const float* __restrict__ q,
                                   const float* __restrict__ k,
                                   const int* __restrict__ ei,   // [2,E]
                                   float* __restrict__ score,    // [E,8] bias in / score out
                                   float* __restrict__ segmax,   // [N,8]
                                   int E) {
    const int e    = blockIdx.x * (blockDim.x >> 5) + (threadIdx.x >> 5);
    const int lane = threadIdx.x & 31;
    if (e >= E) return;
    const int src = ei[e];
    const int dst = ei[E + e];

    const float4 qv = *(const float4*)(q + (size_t)dst * DIMM + lane * 4);
    const float4 kv = *(const float4*)(k + (size_t)src * DIMM + lane * 4);
    float p = qv.x * kv.x + qv.y * kv.y + qv.z * kv.z + qv.w * kv.w;
    p += __shfl_xor(p, 1, 32);
    p += __shfl_xor(p, 2, 32);           // lanes 4h now hold the head-h dot

    if ((lane & 3) == 0) {
        const int h = lane >> 2;
        const float s = p * 0.25f + score[(size_t)e * HH + h];  // 1/sqrt(16)
        score[(size_t)e * HH + h] = s;
        atomicMaxFloat(&segmax[(size_t)dst * HH + h], s);
    }
}

// ---------------------------------------------------------------------------
// ex = exp(score - segmax[dst]); accumulate denominators.
// ---------------------------------------------------------------------------
__global__ void edge_exp_kernel(const int* __restrict__ ei,
                                float* __restrict__ score,       // in: score, out: ex
                                const float* __restrict__ segmax,
                                float* __restrict__ denom,
                                int E) {
    const int i = blockIdx.x * blockDim.x + threadIdx.x;  // over E*H
    if (i >= E * HH) return;
    const int e = i >> 3;
    const int h = i & 7;
    const int dst = ei[E + e];
    const float ex = __expf(score[i] - segmax[(size_t)dst * HH + h]);
    score[i] = ex;
    atomicAdd(&denom[(size_t)dst * HH + h], ex);
}

// ---------------------------------------------------------------------------
// Scatter messages: out[dst] += v[src] * (ex/denom[dst]). Wave per edge,
// lane = (head, 4 channels); 4 fp32 atomics per lane, L2-resident target.
// ---------------------------------------------------------------------------
__global__ void scatter_kernel(const float* __restrict__ v,
                               const int* __restrict__ ei,
                               const float* __restrict__ score,   // ex
                               const float* __restrict__ denom,
                               float* __restrict__ out,
                               int E) {
    const int e    = blockIdx.x * (blockDim.x >> 5) + (threadIdx.x >> 5);
    const int lane = threadIdx.x & 31;
    if (e >= E) return;
    const int src = ei[e];
    const int dst = ei[E + e];
    const int h   = lane >> 2;

    const float alpha = score[(size_t)e * HH + h] / denom[(size_t)dst * HH + h];
    const float4 vv = *(const float4*)(v + (size_t)src * DIMM + lane * 4);
    float* o = out + (size_t)dst * DIMM + lane * 4;
    atomicAdd(o + 0, vv.x * alpha);
    atomicAdd(o + 1, vv.y * alpha);
    atomicAdd(o + 2, vv.z * alpha);
    atomicAdd(o + 3, vv.w * alpha);
}

// ---------------------------------------------------------------------------
extern "C" void kernel_launch(void* const* d_in, const int* in_sizes, int n_in,
                              void* d_out, int out_size, void* d_ws, size_t ws_size,
                              hipStream_t stream) {
    const float* x  = (const float*)d_in[0];   // [N,128]
    const int*   ei = (const int*)d_in[1];     // [2,E]
    const float* ea = (const float*)d_in[2];   // [E,32]
    const float* Wq = (const float*)d_in[3];   // [128,128]
    const float* Wk = (const float*)d_in[4];
    const float* Wv = (const float*)d_in[5];
    const float* We = (const float*)d_in[6];   // [8,32]
    float* out = (float*)d_out;                // [N,128]

    // Workspace layout (floats)
    float* q      = (float*)d_ws;
    float* k      = q + (size_t)NN * DIMM;
    float* v      = k + (size_t)NN * DIMM;
    float* score  = v + (size_t)NN * DIMM;     // [E,8]
    float* segmax = score + (size_t)EE * HH;   // [N,8]
    float* denom  = segmax + (size_t)NN * HH;  // [N,8]

    // 1) init out / segmax / denom
    {
        const int nOut = NN * DIMM, nSeg = NN * HH;
        init_kernel<<<(nOut + 255) / 256, 256, 0, stream>>>(out, segmax, denom, nOut, nSeg);
    }

    // 2) WMMA GEMMs: q,k,v = x*W^T ; bias = edge_attr*We^T (into score buf)
    {
        const int tilesQKV = (NN / 16) * (DIMM / 16);          // 50000 tiles
        const int blocks   = (tilesQKV + 7) / 8;               // 8 waves/block
        gemm_xwT_wmma<DIMM, DIMM, DIMM><<<blocks, 256, 0, stream>>>(x, Wq, q, NN);
        gemm_xwT_wmma<DIMM, DIMM, DIMM><<<blocks, 256, 0, stream>>>(x, Wk, k, NN);
        gemm_xwT_wmma<DIMM, DIMM, DIMM><<<blocks, 256, 0, stream>>>(x, Wv, v, NN);

        const int tilesB = EE / 16;                            // 100000 tiles
        gemm_xwT_wmma<EDIM, HH, HH><<<(tilesB + 7) / 8, 256, 0, stream>>>(ea, We, score, EE);
    }

    // 3) per-edge scores + segment max (8 waves / 256-thread block)
    edge_scores_kernel<<<EE / 8, 256, 0, stream>>>(q, k, ei, score, segmax, EE);

    // 4) exponentials + denominators
    edge_exp_kernel<<<(EE * HH + 255) / 256, 256, 0, stream>>>(ei, score, segmax, denom, EE);

    // 5) weighted scatter-add of values
    scatter_kernel<<<EE / 8, 256, 0, stream>>>(v, ei, score, denom, out, EE);
}